// ModuleCorrelationTranspose_82910048682529
// MI455X (gfx1250) — compile-verified
//
#include <hip/hip_runtime.h>

typedef __attribute__((ext_vector_type(2))) float v2f;
typedef __attribute__((ext_vector_type(8))) float v8f;

#define MDv    4
#define DDv    9      // window size (9x9)
#define K81    81
#define Bv     4
#define Cv     128
#define Hv     192
#define Wv     192
#define TW     16     // output w-tile per block
#define XW     24     // x span per tile (TW + 2*MD)
#define XWP    28     // padded stride: rows 8B-aligned, bank-conflict-free
#define PFROWS 40     // 16 guard + 9 valid + 15 guard rows per pr
#define NLOAD  12     // (Cv*XW)/256 second-elements per thread

__global__ __launch_bounds__(256) void corr_transpose_wmma(
    const float* __restrict__ first,
    const float* __restrict__ second,
    float* __restrict__ out)
{
    // Zero-guarded Toeplitz source: pf[pr][16+d][n] = first[b, pr*9+d, h, w0+n]
    // for 0<=d<=8, zero for all other d in [-16, 23].
    __shared__ __align__(16) float pf[DDv][PFROWS][TW];
    // Double-buffered second row tile.
    __shared__ __align__(16) float sec[2][Cv][XWP];

    const int wt  = blockIdx.x;           // 0..11
    const int h   = blockIdx.y;           // 0..191
    const int b   = blockIdx.z;           // 0..3
    const int w0  = wt * TW;
    const int tid = threadIdx.x;

    // ---- fill pf (zeros + data in one pass, disjoint per thread) ----
    const float* fbase = first + ((size_t)(b * K81) * Hv + h) * Wv + w0;
    for (int i = tid; i < DDv * PFROWS * TW; i += 256) {
        const int n  = i & 15;
        const int t  = i >> 4;            // 0..359
        const int pr = t / PFROWS;
        const int d  = (t - pr * PFROWS) - 16;
        float v = 0.0f;
        if ((unsigned)d <= 8u)
            v = fbase[(size_t)(pr * DDv + d) * (Hv * Wv) + n];
        pf[pr][d + 16][n] = v;
    }

    // ---- precompute the 12 loop-invariant second-load slots ----
    const float* sbase = second + (size_t)b * Cv * Hv * Wv;
    int  goff[NLOAD];   // global element offset (row 0, x clamped)
    int  loff[NLOAD];   // LDS element offset within one sec buffer
    bool okx [NLOAD];   // x in range?
    #pragma unroll
    for (int j = 0; j < NLOAD; ++j) {
        const int i  = tid + 256 * j;     // 0..3071
        const int x  = i % XW;
        const int c  = i / XW;
        const int gx = w0 - MDv + x;
        okx[j]  = (unsigned)gx < (unsigned)Wv;
        const int gxc = min(max(gx, 0), Wv - 1);
        goff[j] = c * (Hv * Wv) + gxc;
        loff[j] = c * XWP + x;
    }
    float* secf = &sec[0][0][0];

    // ---- preload row pr=0 into buffer 0 (branchless: clamp + select) ----
    {
        const int  row  = h - MDv;
        const bool okr  = (unsigned)row < (unsigned)Hv;
        const int  rowc = min(max(row, 0), Hv - 1);
        #pragma unroll
        for (int j = 0; j < NLOAD; ++j) {
            const float v = sbase[goff[j] + rowc * Wv];
            secf[loff[j]] = (okx[j] && okr) ? v : 0.0f;
        }
    }
    __syncthreads();

    const int lane = tid & 31;
    const int wave = tid >> 5;
    const int c0   = wave * 16;           // 8 waves cover all 128 channels
    const int mn   = lane & 15;           // M(=c) for A, N(=w) for B/C/D
    const int half = lane >> 4;

    v8f acc = {0.f, 0.f, 0.f, 0.f, 0.f, 0.f, 0.f, 0.f};

    for (int pr = 0; pr < DDv; ++pr) {
        const int cur = pr & 1;

        // ---- prefetch next second row into the other buffer ----
        if (pr + 1 < DDv) {
            const int  row  = h - MDv + (pr + 1);
            const bool okr  = (unsigned)row < (unsigned)Hv;
            const int  rowc = min(max(row, 0), Hv - 1);
            float* dst = secf + ((pr + 1) & 1) * (Cv * XWP);
            #pragma unroll
            for (int j = 0; j < NLOAD; ++j) {
                const float v = sbase[goff[j] + rowc * Wv];
                dst[loff[j]] = (okx[j] && okr) ? v : 0.0f;
            }
        }

        // ---- GEMM: out[c16, w16] += A(16xK) * ZeroPaddedToeplitz(first)(Kx16)
        const float* srow = secf + cur * (Cv * XWP) + (c0 + mn) * XWP;
        #pragma unroll
        for (int kc = 0; kc < 6; ++kc) {
            const int kg0 = kc * 4 + 2 * half;       // K of element 0, this lane

            // A fragment: one ds_load_b64 (8B-aligned, conflict-free)
            const v2f a = *(const v2f*)(srow + kg0);

            // B fragment: unconditional reads from zero-guarded pf
            const int d0 = kg0 - mn;                 // in [-15, 22]
            v2f bf;
            bf.x = pf[pr][16 + d0][mn];
            bf.y = pf[pr][17 + d0][mn];

            acc = __builtin_amdgcn_wmma_f32_16x16x4_f32(
                /*neg_a=*/false, a, /*neg_b=*/false, bf,
                /*c_mod=*/(short)0, acc, /*reuse_a=*/false, /*reuse_b=*/false);
        }
        __syncthreads();  // next buffer complete; current buffer reusable
    }

    // ---- writeback: C/D layout vgpr j -> M = j + 8*half, N = mn ----
    const float scale = 1.0f / (float)Cv;
    float* obase = out + ((size_t)(b * Cv) * Hv + h) * Wv;
    #pragma unroll
    for (int j = 0; j < 8; ++j) {
        const int c = c0 + j + 8 * half;
        obase[(size_t)c * (Hv * Wv) + w0 + mn] = acc[j] * scale;
    }
}

extern "C" void kernel_launch(void* const* d_in, const int* in_sizes, int n_in,
                              void* d_out, int out_size, void* d_ws, size_t ws_size,
                              hipStream_t stream) {
    const float* first  = (const float*)d_in[0];   // (4, 81, 192, 192) fp32
    const float* second = (const float*)d_in[1];   // (4, 128, 192, 192) fp32
    float* out = (float*)d_out;                    // (4, 128, 192, 192) fp32

    dim3 grid(Wv / TW, Hv, Bv);   // (12, 192, 4)
    corr_transpose_wmma<<<grid, 256, 0, stream>>>(first, second, out);
}